// FCNNPerceptionModule_33646773797511
// MI455X (gfx1250) — compile-verified
//
#include <hip/hip_runtime.h>

// ---------------- problem constants (match reference) ----------------
#define BATCH   16
#define NANCH   25200
#define NCLS    80
#define PSTRIDE 85          // 4 box + 1 obj + 80 cls
#define NCAND   1024
#define NBINS   4096
#define CONF_T  0.25f
#define IOU_T   0.45f

typedef __attribute__((ext_vector_type(2))) float v2f;
typedef __attribute__((ext_vector_type(8))) float v8f;

// =====================================================================
// Kernel 1: per-anchor confidence.  score = obj * max(cls), argmax cls.
// Memory-bound sweep over the 137MB input (fits MI455X 192MB L2).
// =====================================================================
__global__ void k_scores(const float* __restrict__ pred,
                         float* __restrict__ scores,
                         int*   __restrict__ clsid) {
    int gid = blockIdx.x * blockDim.x + threadIdx.x;
    if (gid >= BATCH * NANCH) return;
    const float* p = pred + (size_t)gid * PSTRIDE;
    float obj = p[4];
    float best = -1.0f; int bi = 0;
#pragma unroll 8
    for (int j = 0; j < NCLS; ++j) {
        float v = p[5 + j];
        if (v > best) { best = v; bi = j; }
    }
    float s = obj * best;
    bool valid = (obj > CONF_T) && (s > CONF_T);
    scores[gid] = valid ? s : 0.0f;
    clsid[gid]  = bi;
}

// =====================================================================
// Kernel 2: per-batch top-1024 select (LDS histogram radix-select) +
// in-LDS bitonic sort on packed u64 key (score desc, index asc).
// One 1024-thread workgroup per batch.
// =====================================================================
__global__ void k_select(const float* __restrict__ scores,
                         float* __restrict__ sel_score,
                         int*   __restrict__ sel_idx) {
    __shared__ unsigned int hist[NBINS];
    __shared__ unsigned long long keys[NCAND];
    __shared__ unsigned int s_cnt;
    __shared__ int s_T;

    const int b   = blockIdx.x;
    const int tid = threadIdx.x;
    const float* sc = scores + (size_t)b * NANCH;

    for (int i = tid; i < NBINS; i += 1024) hist[i] = 0u;
    if (tid == 0) s_cnt = 0u;
    __syncthreads();

    // histogram of positive (already thresholded) scores, 4096 bins on [0,1]
    for (int a = tid; a < NANCH; a += 1024) {
        float s = sc[a];
        if (s > 0.0f) {
            int k = (int)(s * 4096.0f);
            k = (k > 4095) ? 4095 : (k < 1 ? 1 : k);
            atomicAdd(&hist[k], 1u);
        }
    }
    __syncthreads();

    // threshold bin: largest T with count(bin >= T) >= NCAND
    if (tid == 0) {
        unsigned int cum = 0; int T = 0;
        for (int bn = NBINS - 1; bn >= 1; --bn) {
            cum += hist[bn];
            if (cum >= NCAND) { T = bn; break; }
        }
        s_T = T;
    }
    for (int i = tid; i < NCAND; i += 1024)
        keys[i] = 0x00000000FFFFFFFFull;     // pad: score=0, idx=0 (via ~idx)
    __syncthreads();
    const int T = s_T;

    // pass 1: bins strictly above threshold (guaranteed < NCAND of them)
    for (int a = tid; a < NANCH; a += 1024) {
        float s = sc[a];
        if (s > 0.0f) {
            int k = (int)(s * 4096.0f);
            k = (k > 4095) ? 4095 : (k < 1 ? 1 : k);
            if (k > T) {
                unsigned int pos = atomicAdd(&s_cnt, 1u);
                if (pos < NCAND)
                    keys[pos] = ((unsigned long long)__float_as_uint(s) << 32)
                              | (unsigned int)(~a);
            }
        }
    }
    __syncthreads();
    // pass 2: fill remaining slots from the threshold bin
    for (int a = tid; a < NANCH; a += 1024) {
        float s = sc[a];
        if (s > 0.0f) {
            int k = (int)(s * 4096.0f);
            k = (k > 4095) ? 4095 : (k < 1 ? 1 : k);
            if (k == T) {
                unsigned int pos = atomicAdd(&s_cnt, 1u);
                if (pos < NCAND)
                    keys[pos] = ((unsigned long long)__float_as_uint(s) << 32)
                              | (unsigned int)(~a);
            }
        }
    }
    __syncthreads();

    // bitonic sort, descending (score major, ~idx minor -> ties by low index)
    for (unsigned int k = 2; k <= NCAND; k <<= 1) {
        for (unsigned int j = k >> 1; j > 0; j >>= 1) {
            unsigned int ixj = tid ^ j;
            if (ixj > (unsigned int)tid) {
                unsigned long long x = keys[tid], y = keys[ixj];
                bool up = ((tid & k) == 0);
                if (up ? (x < y) : (x > y)) { keys[tid] = y; keys[ixj] = x; }
            }
            __syncthreads();
        }
    }

    unsigned long long kk = keys[tid];
    sel_score[b * NCAND + tid] = __uint_as_float((unsigned int)(kk >> 32));
    unsigned int ia = ~(unsigned int)(kk & 0xFFFFFFFFu);
    sel_idx[b * NCAND + tid] = (ia >= NANCH) ? 0 : (int)ia;
}

// =====================================================================
// Kernel 3: NMS. One 1024-thread workgroup per batch.
//  - candidate boxes/areas/scores/classes staged in LDS
//  - suppression bitmask built tile-wise; the IoU union term
//    area_a[i] + area_b[j] + 1e-9  is computed entirely by the matrix
//    unit:  V_WMMA_F32_16X16X4_F32 with A=[area|1|0|0] (16x4),
//    B=[1;area;0;0] (4x16), C=1e-9  ->  D = A*B + C  (exact f32)
//  - iou > t  rewritten division-free:  inter > t*(sum - inter)
//  - per-tile halfword mask ownership -> plain b16 stores, no atomics
//  - greedy suppression: one wave, 32-word keep mask, mask rows from L2
//  - output: sorted kept prefix = top-e rows
// =====================================================================
__global__ void k_nms(const float* __restrict__ pred,
                      const int*   __restrict__ clsid,
                      const float* __restrict__ sel_score,
                      const int*   __restrict__ sel_idx,
                      unsigned int* __restrict__ mask,   // [BATCH][NCAND][32]
                      float* __restrict__ out, int e) {
    __shared__ float bx1[NCAND], by1[NCAND], bx2[NCAND], by2[NCAND];
    __shared__ float barea[NCAND], bscore[NCAND];
    __shared__ int   bcls[NCAND];
    __shared__ unsigned int keep[32];

    const int b   = blockIdx.x;
    const int tid = threadIdx.x;
    unsigned int*   maskb   = mask + (size_t)b * NCAND * 32;
    unsigned short* maskb16 = (unsigned short*)maskb;      // 64 halfwords/row
    float* outb = out + (size_t)b * e * 6;

    // stage candidate tid
    {
        int idx = sel_idx[b * NCAND + tid];
        const float* p = pred + ((size_t)b * NANCH + idx) * PSTRIDE;
        float cx = p[0], cy = p[1], w = p[2], h = p[3];
        bx1[tid] = cx - 0.5f * w;  by1[tid] = cy - 0.5f * h;
        bx2[tid] = cx + 0.5f * w;  by2[tid] = cy + 0.5f * h;
        barea[tid]  = w * h;
        bscore[tid] = sel_score[b * NCAND + tid];
        bcls[tid]   = clsid[b * NANCH + idx];
    }
    // zero this batch's mask + output section
    for (int i = tid; i < NCAND * 32; i += 1024) maskb[i] = 0u;
    for (int i = tid; i < e * 6; i += 1024) outb[i] = 0.0f;
    if (tid < 32) keep[tid] = 0xFFFFFFFFu;
    __syncthreads();

    // ---- tile phase: 64x64 grid of 16x16 tiles, upper triangle ----
    const int wave = tid >> 5;
    const int lane = tid & 31;
    const int NT = NCAND / 16;              // 64
    const int TOT = NT * (NT + 1) / 2;      // 2080
    for (int t = wave; t < TOT; t += 32) {
        int ti = 0, rem = t;
        while (rem >= NT - ti) { rem -= NT - ti; ++ti; }
        int tj = ti + rem;
        int i0 = ti * 16, j0 = tj * 16;

        // D[m][n] = area_a[i0+m] + area_b[j0+n] + 1e-9 (eps via accumulator)
        float aA = barea[i0 + (lane & 15)];
        float aB = barea[j0 + (lane & 15)];
        v2f Aop; Aop[0] = (lane < 16) ? aA   : 0.0f;  Aop[1] = (lane < 16) ? 1.0f : 0.0f;
        v2f Bop; Bop[0] = (lane < 16) ? 1.0f : 0.0f;  Bop[1] = (lane < 16) ? aB   : 0.0f;
        v8f Ceps = {1e-9f, 1e-9f, 1e-9f, 1e-9f, 1e-9f, 1e-9f, 1e-9f, 1e-9f};
        v8f D = __builtin_amdgcn_wmma_f32_16x16x4_f32(
                    false, Aop, false, Bop, (short)0, Ceps, false, false);

        const int n = lane & 15;
        const int mhi = (lane >> 4) << 3;   // 0 for lanes 0-15, 8 for 16-31
#pragma unroll
        for (int k = 0; k < 8; ++k) {
            int gi = i0 + mhi + k;
            int gj = j0 + n;
            float xx1 = fmaxf(bx1[gi], bx1[gj]);
            float yy1 = fmaxf(by1[gi], by1[gj]);
            float xx2 = fminf(bx2[gi], bx2[gj]);
            float yy2 = fminf(by2[gi], by2[gj]);
            float iw = fmaxf(xx2 - xx1, 0.0f);
            float ih = fmaxf(yy2 - yy1, 0.0f);
            // class offset (+cls*7680) cancels in-class; cross-class iou==0
            float inter = (bcls[gi] == bcls[gj]) ? (iw * ih) : 0.0f;
            // iou > t  <=>  inter > t*(union),  union = D[k]-inter (+eps) > 0
            bool sup = (inter > IOU_T * (D[k] - inter)) && (gj > gi);
            unsigned long long bal = __ballot(sup);      // wave32: low 32 bits
            if (lane == 0) {
                unsigned int lo = (unsigned int)(bal & 0xFFFFu);          // rows i0+k
                unsigned int hi = (unsigned int)((bal >> 16) & 0xFFFFu);  // rows i0+8+k
                int hw = j0 >> 4;           // halfword column index (0..63)
                // each (row, halfword) written by exactly one tile: plain b16
                if (lo) maskb16[(size_t)(i0 + k)     * 64 + hw] = (unsigned short)lo;
                if (hi) maskb16[(size_t)(i0 + 8 + k) * 64 + hw] = (unsigned short)hi;
            }
        }
    }
    __syncthreads();

    // ---- greedy sequential suppression: one wave, keep mask in LDS ----
    if (tid < 32) {
        for (int i = 0; i < NCAND; ++i) {
            unsigned int kw = keep[i >> 5];
            bool live = ((kw >> (i & 31)) & 1u) && (bscore[i] > 0.0f);
            if (live) {
                unsigned int mrow = maskb[(size_t)i * 32 + tid];
                if (mrow) keep[tid] &= ~mrow;   // row i only has bits j>i
            }
        }
    }
    __syncthreads();

    // ---- output: scores still sorted desc, so kept prefix == top-e ----
    if (tid == 0) {
        int cnt = 0;
        for (int c = 0; c < NCAND && cnt < e; ++c) {
            if (((keep[c >> 5] >> (c & 31)) & 1u) && bscore[c] > 0.0f) {
                float* r = outb + (size_t)cnt * 6;
                r[0] = bx1[c]; r[1] = by1[c]; r[2] = bx2[c]; r[3] = by2[c];
                r[4] = bscore[c]; r[5] = (float)bcls[c];
                ++cnt;
            }
        }
    }
}

// =====================================================================
extern "C" void kernel_launch(void* const* d_in, const int* in_sizes, int n_in,
                              void* d_out, int out_size, void* d_ws, size_t ws_size,
                              hipStream_t stream) {
    const float* pred = (const float*)d_in[0];
    const int e = out_size / (BATCH * 6);    // reference e (=50)

    char* ws = (char*)d_ws;
    size_t off = 0;
    float* scores   = (float*)(ws + off); off += (size_t)BATCH * NANCH * sizeof(float);
    int*   clsid    = (int*)  (ws + off); off += (size_t)BATCH * NANCH * sizeof(int);
    float* selScore = (float*)(ws + off); off += (size_t)BATCH * NCAND * sizeof(float);
    int*   selIdx   = (int*)  (ws + off); off += (size_t)BATCH * NCAND * sizeof(int);
    unsigned int* mask = (unsigned int*)(ws + off);
    // mask uses BATCH*NCAND*32*4 = 2MB; total ws ~5.5MB

    const int total = BATCH * NANCH;
    k_scores<<<(total + 255) / 256, 256, 0, stream>>>(pred, scores, clsid);
    k_select<<<BATCH, 1024, 0, stream>>>(scores, selScore, selIdx);
    k_nms   <<<BATCH, 1024, 0, stream>>>(pred, clsid, selScore, selIdx, mask,
                                         (float*)d_out, e);
}